// DLDetFPNHead_49709951484028
// MI455X (gfx1250) — compile-verified
//
#include <hip/hip_runtime.h>
#include <hip/hip_bf16.h>
#include <stdint.h>

typedef __attribute__((ext_vector_type(2))) float v2f;
typedef __attribute__((ext_vector_type(8))) float v8f;

#define NB 2
#define KB 32
#define CB 80
#define HWPIX 262144   // 512*512
#define CHUNKS 64
#define T1 256

// ---------------------------------------------------------------------------
// Kernel 0: zero the 3*N*K f32 accumulators in workspace.
// ---------------------------------------------------------------------------
__global__ __launch_bounds__(256) void zero_ws(float* __restrict__ ws) {
    int i = threadIdx.x;
    if (i < 3 * NB * KB) ws[i] = 0.0f;
}

// ---------------------------------------------------------------------------
// Kernel 1: masked count / sum / sum-of-squares per (n,k).
// grid = (CHUNKS, N*K), 256 threads. float4 emb loads + 4-packed mask bytes.
// Bandwidth-bound: ~80 MB total traffic.
// ---------------------------------------------------------------------------
__global__ __launch_bounds__(T1) void masked_sums(const float*   __restrict__ emb,
                                                  const uint8_t* __restrict__ mask,
                                                  const int*     __restrict__ cls,
                                                  float*         __restrict__ ws) {
    const int pair = blockIdx.y;              // n*KB + k
    const int n    = pair / KB;
    const int c    = cls[pair];               // uniform per block (scalar load)

    const float4*   e4 = (const float4*)(emb + ((size_t)n * CB + (size_t)c) * (size_t)HWPIX);
    const uint32_t* m4 = (const uint32_t*)(mask + (size_t)pair * (size_t)HWPIX);

    float s = 0.0f, s2 = 0.0f, cn = 0.0f;
    const int nq = HWPIX / 4;
    for (int q = blockIdx.x * T1 + threadIdx.x; q < nq; q += CHUNKS * T1) {
        float4   e = e4[q];
        uint32_t m = m4[q];
        if (m & 0x000000FFu) { s += e.x; s2 += e.x * e.x; cn += 1.0f; }
        if (m & 0x0000FF00u) { s += e.y; s2 += e.y * e.y; cn += 1.0f; }
        if (m & 0x00FF0000u) { s += e.z; s2 += e.z * e.z; cn += 1.0f; }
        if (m & 0xFF000000u) { s += e.w; s2 += e.w * e.w; cn += 1.0f; }
    }

    // wave32 reduction
    for (int off = 16; off > 0; off >>= 1) {
        s  += __shfl_down(s,  off, 32);
        s2 += __shfl_down(s2, off, 32);
        cn += __shfl_down(cn, off, 32);
    }
    __shared__ float rs[T1 / 32], rs2[T1 / 32], rc[T1 / 32];
    const int wid = threadIdx.x >> 5;
    if ((threadIdx.x & 31) == 0) { rs[wid] = s; rs2[wid] = s2; rc[wid] = cn; }
    __syncthreads();
    if (threadIdx.x == 0) {
        float S = 0.0f, S2 = 0.0f, C = 0.0f;
        for (int w = 0; w < T1 / 32; ++w) { S += rs[w]; S2 += rs2[w]; C += rc[w]; }
        atomicAdd(&ws[pair],               S);
        atomicAdd(&ws[NB * KB + pair],     S2);
        atomicAdd(&ws[2 * NB * KB + pair], C);
    }
}

// ---------------------------------------------------------------------------
// Kernel 2: single-wave epilogue. Lane k owns instance k (K==32==wave32).
// The 32x32 mean outer product for the pairwise hinge term is computed with
// four V_WMMA_F32_16X16X4_F32 tiles (only K=0 of A/B populated, rest zero).
// Block = exactly one wave, no divergent branches -> EXEC all-ones for WMMA.
// ---------------------------------------------------------------------------
__global__ __launch_bounds__(32) void finalize(const float* __restrict__ ws,
                                               const int*   __restrict__ cls,
                                               float*       __restrict__ out) {
    const int lane = threadIdx.x;             // == k
    __shared__ float sMean[KB];
    __shared__ int   sCls[KB];

    float total = 0.0f;
    for (int n = 0; n < NB; ++n) {
        const int pair = n * KB + lane;
        const float s    = ws[pair];
        const float s2   = ws[NB * KB + pair];
        const float cn   = ws[2 * NB * KB + pair];
        const bool  valid = cn > 0.0f;
        const float safe  = valid ? cn : 1.0f;
        const float mean  = valid ? (s / safe) : 0.0f;
        const float var   = valid ? (s2 / safe - mean * mean) : 0.0f;

        sMean[lane] = mean;
        sCls[lane]  = cls[pair];
        __syncthreads();

        // reg = mean(mean^2), intra = mean(var)  (wave32 xor-reduce)
        float reg = mean * mean, intra = var;
        for (int off = 16; off > 0; off >>= 1) {
            reg   += __shfl_xor(reg,   off, 32);
            intra += __shfl_xor(intra, off, 32);
        }
        reg   *= (1.0f / (float)KB);
        intra *= (1.0f / (float)KB);

        // inter: mean outer product via WMMA (2x2 tiles of 16x16)
        float inter = 0.0f;
        for (int ti = 0; ti < 2; ++ti) {
            for (int tj = 0; tj < 2; ++tj) {
                // A[m][0] = mean[16*ti+m]; all other K columns zero.
                // B[0][nn] = mean[16*tj+nn]; all other K rows zero.
                const float a0 = (lane < 16) ? sMean[ti * 16 + lane] : 0.0f;
                const float b0 = (lane < 16) ? sMean[tj * 16 + lane] : 0.0f;
                v2f A = {a0, 0.0f};
                v2f B = {b0, 0.0f};
                v8f Cm = {};
                // D[m][nn] = mean[16*ti+m] * mean[16*tj+nn]
                v8f D = __builtin_amdgcn_wmma_f32_16x16x4_f32(
                    /*neg_a=*/false, A, /*neg_b=*/false, B,
                    /*c_mod=*/(short)0, Cm, /*reuse_a=*/false, /*reuse_b=*/false);

                const int Nloc  = (lane < 16) ? lane : (lane - 16);
                const int Mbase = (lane < 16) ? 0 : 8;
                const int kcol  = tj * 16 + Nloc;
                const float mk  = sMean[kcol];
                const int   ck  = sCls[kcol];
#pragma unroll
                for (int v = 0; v < 8; ++v) {
                    const int   jrow = ti * 16 + Mbase + v;
                    const float mj   = sMean[jrow];
                    const float d2   = mj * mj + mk * mk - 2.0f * D[v];
                    if (jrow < kcol && sCls[jrow] == ck)
                        inter += fmaxf(1.0f - d2, 0.0f);
                }
            }
        }
        for (int off = 16; off > 0; off >>= 1) inter += __shfl_xor(inter, off, 32);

        total += inter + reg + intra;
        __syncthreads();   // before next image overwrites sMean/sCls
    }

    if (lane == 0) out[0] = total * (1.0f / (float)NB) * 0.1f;
}

// ---------------------------------------------------------------------------
extern "C" void kernel_launch(void* const* d_in, const int* in_sizes, int n_in,
                              void* d_out, int out_size, void* d_ws, size_t ws_size,
                              hipStream_t stream) {
    const float*   emb  = (const float*)d_in[0];     // (N,C,H,W) f32
    const uint8_t* mask = (const uint8_t*)d_in[1];   // (N,K,H,W) bool (1 byte)
    const int*     cls  = (const int*)d_in[2];       // (N,K) int32
    float*         ws   = (float*)d_ws;              // 3*N*K f32 accumulators
    float*         out  = (float*)d_out;

    zero_ws<<<1, 256, 0, stream>>>(ws);
    dim3 grid(CHUNKS, NB * KB);
    masked_sums<<<grid, T1, 0, stream>>>(emb, mask, cls, ws);
    finalize<<<1, 32, 0, stream>>>(ws, cls, out);
}